// GraphSAGE_67551245631640
// MI455X (gfx1250) — compile-verified
//
#include <hip/hip_runtime.h>
#include <math.h>

typedef float v2f __attribute__((ext_vector_type(2)));
typedef float v8f __attribute__((ext_vector_type(8)));

#define NN 50000
#define NE 800000
#define FEAT 128
#define NG 256
#define MTILES (NN / 16)          // 3125 (divides exactly)
#define NTILES (FEAT / 16)        // 8
#define NWAVES (MTILES * NTILES)  // 25000

__device__ __forceinline__ void atomAdd(float* p, float v) {
    unsafeAtomicAdd(p, v);  // hardware global_atomic_add_f32
}

// ---- count occurrences of idx[i] (degree / per-graph node count) ----
__global__ void count_kernel(const long long* __restrict__ idx, int n,
                             float* __restrict__ out) {
    int i = blockIdx.x * blockDim.x + threadIdx.x;
    if (i < n) atomAdd(&out[(int)idx[i]], 1.0f);
}

// ---- one wave per edge: agg[dst] += feat[src] (128 feats = 32 lanes x float4) ----
__global__ __launch_bounds__(256)
void scatter_kernel(const float* __restrict__ feat,
                    const long long* __restrict__ src,
                    const long long* __restrict__ dst,
                    float* __restrict__ agg) {
    int wave = (blockIdx.x * blockDim.x + threadIdx.x) >> 5;
    int lane = threadIdx.x & 31;
    if (wave >= NE) return;
    int s = (int)src[wave];
    int d = (int)dst[wave];
    const float4 v = *(const float4*)(feat + (size_t)s * FEAT + lane * 4);
    float* p = agg + (size_t)d * FEAT + lane * 4;
    atomAdd(p + 0, v.x);
    atomAdd(p + 1, v.y);
    atomAdd(p + 2, v.z);
    atomAdd(p + 3, v.w);
}

// ---- one wave per node: pooled[batch[n]] += h[n] ----
__global__ __launch_bounds__(256)
void pool_kernel(const float* __restrict__ h,
                 const long long* __restrict__ batch,
                 float* __restrict__ pooled) {
    int wave = (blockIdx.x * blockDim.x + threadIdx.x) >> 5;
    int lane = threadIdx.x & 31;
    if (wave >= NN) return;
    int g = (int)batch[wave];
    const float4 v = *(const float4*)(h + (size_t)wave * FEAT + lane * 4);
    float* p = pooled + (size_t)g * FEAT + lane * 4;
    atomAdd(p + 0, v.x);
    atomAdd(p + 1, v.y);
    atomAdd(p + 2, v.z);
    atomAdd(p + 3, v.w);
}

// ---- fused: out = relu( (agg/deg) @ Wl + X @ Wr + b ), fp32 WMMA 16x16x4 ----
// One wave computes one 16x16 output tile; K=128 -> 32+32 WMMA steps.
__global__ __launch_bounds__(128)
void sage_gemm_relu(const float* __restrict__ Aagg,  // [NN,128] neighbor sums
                    const float* __restrict__ X,     // [NN,128] self features
                    const float* __restrict__ Wl,    // [128,128] row-major
                    const float* __restrict__ Wr,    // [128,128]
                    const float* __restrict__ bias,  // [128]
                    const float* __restrict__ deg,   // [NN]
                    float* __restrict__ out) {       // [NN,128]
    const int wave = blockIdx.x * (blockDim.x >> 5) + (threadIdx.x >> 5);
    const int lane = threadIdx.x & 31;
    if (wave >= NWAVES) return;
    const int m0 = (wave >> 3) * 16;
    const int n0 = (wave & 7) * 16;
    const int half = lane >> 4;       // 0: K even pair, 1: K odd pair (offset +2)
    const int l = lane & 15;
    const int rowA = m0 + l;          // A-frag: both elements share this row
    const int colB = n0 + l;          // B-frag / C-frag column
    const float rscale = 1.0f / fmaxf(deg[rowA], 1.0f); // fold mean into A load

    const float* Ar = Aagg + (size_t)rowA * FEAT + 2 * half;
    const float* Xr = X + (size_t)rowA * FEAT + 2 * half;

    v8f acc = {};  // C starts at 0; bias added after
#pragma unroll 8
    for (int k = 0; k < FEAT; k += 4) {
        const int kk = k + 2 * half;
        v2f a = *(const v2f*)(Ar + k);        // A[rowA][kk], A[rowA][kk+1]
        a.x *= rscale;
        a.y *= rscale;
        v2f b;
        b.x = Wl[kk * FEAT + colB];           // B[kk][colB]
        b.y = Wl[(kk + 1) * FEAT + colB];     // B[kk+1][colB]
        acc = __builtin_amdgcn_wmma_f32_16x16x4_f32(
            false, a, false, b, (short)0, acc, false, false);
    }
#pragma unroll 8
    for (int k = 0; k < FEAT; k += 4) {
        const int kk = k + 2 * half;
        v2f a = *(const v2f*)(Xr + k);
        v2f b;
        b.x = Wr[kk * FEAT + colB];
        b.y = Wr[(kk + 1) * FEAT + colB];
        acc = __builtin_amdgcn_wmma_f32_16x16x4_f32(
            false, a, false, b, (short)0, acc, false, false);
    }

    const float bv = bias[colB];
#pragma unroll
    for (int j = 0; j < 8; ++j) {
        // C/D layout: VGPR j -> row m0 + j + 8*half, col n0 + l
        float v = fmaxf(acc[j] + bv, 0.0f);
        out[(size_t)(m0 + j + 8 * half) * FEAT + colB] = v;
    }
}

// ---- head: logits = (pooled/cnt) @ Wfc + bfc; log_softmax over 2 classes ----
__global__ void head_kernel(const float* __restrict__ pooled,
                            const float* __restrict__ cnt,
                            const float* __restrict__ Wfc,  // [128,2]
                            const float* __restrict__ bfc,  // [2]
                            float* __restrict__ out) {      // [256,2]
    int g = blockIdx.x * blockDim.x + threadIdx.x;
    if (g >= NG) return;
    float rc = 1.0f / fmaxf(cnt[g], 1.0f);
    float l0 = bfc[0], l1 = bfc[1];
#pragma unroll 4
    for (int f = 0; f < FEAT; ++f) {
        float p = pooled[g * FEAT + f] * rc;
        l0 = fmaf(p, Wfc[f * 2 + 0], l0);
        l1 = fmaf(p, Wfc[f * 2 + 1], l1);
    }
    float m = fmaxf(l0, l1);
    float lse = m + logf(expf(l0 - m) + expf(l1 - m));
    out[g * 2 + 0] = l0 - lse;
    out[g * 2 + 1] = l1 - lse;
}

extern "C" void kernel_launch(void* const* d_in, const int* in_sizes, int n_in,
                              void* d_out, int out_size, void* d_ws, size_t ws_size,
                              hipStream_t stream) {
    const float* x        = (const float*)d_in[0];
    const long long* ei   = (const long long*)d_in[1];
    const long long* batch= (const long long*)d_in[2];
    const float* Wl1 = (const float*)d_in[3];
    const float* Wr1 = (const float*)d_in[4];
    const float* b1  = (const float*)d_in[5];
    const float* Wl2 = (const float*)d_in[6];
    const float* Wr2 = (const float*)d_in[7];
    const float* b2  = (const float*)d_in[8];
    const float* Wfc = (const float*)d_in[9];
    const float* bfc = (const float*)d_in[10];
    const long long* src = ei;            // edge_index[0]
    const long long* dst = ei + NE;       // edge_index[1]

    char* ws = (char*)d_ws;
    size_t off = 0;
    auto alloc = [&](size_t bytes) -> void* {
        off = (off + 255) & ~(size_t)255;
        void* p = ws + off;
        off += bytes;
        return p;
    };
    float* h1     = (float*)alloc((size_t)NN * FEAT * 4);
    float* h2     = (float*)alloc((size_t)NN * FEAT * 4);
    float* agg    = (float*)alloc((size_t)NN * FEAT * 4);
    float* deg    = (float*)alloc((size_t)NN * 4);
    float* cnt    = (float*)alloc((size_t)NG * 4);
    float* pooled = (float*)alloc((size_t)NG * FEAT * 4);

    hipMemsetAsync(deg, 0, (size_t)NN * 4, stream);
    hipMemsetAsync(cnt, 0, (size_t)NG * 4, stream);
    hipMemsetAsync(pooled, 0, (size_t)NG * FEAT * 4, stream);
    hipMemsetAsync(agg, 0, (size_t)NN * FEAT * 4, stream);

    count_kernel<<<(NE + 255) / 256, 256, 0, stream>>>(dst, NE, deg);
    count_kernel<<<(NN + 255) / 256, 256, 0, stream>>>(batch, NN, cnt);

    const int scatter_blocks = (NE * 32 + 255) / 256;   // one wave32 per edge
    const int gemm_blocks = (NWAVES + 3) / 4;           // 4 waves per block

    // layer 1
    scatter_kernel<<<scatter_blocks, 256, 0, stream>>>(x, src, dst, agg);
    sage_gemm_relu<<<gemm_blocks, 128, 0, stream>>>(agg, x, Wl1, Wr1, b1, deg, h1);

    // layer 2 (re-zero agg; stream order makes this safe)
    hipMemsetAsync(agg, 0, (size_t)NN * FEAT * 4, stream);
    scatter_kernel<<<scatter_blocks, 256, 0, stream>>>(h1, src, dst, agg);
    sage_gemm_relu<<<gemm_blocks, 128, 0, stream>>>(agg, h1, Wl2, Wr2, b2, deg, h2);

    // pool + head
    pool_kernel<<<(NN * 32 + 255) / 256, 256, 0, stream>>>(h2, batch, pooled);
    head_kernel<<<1, 256, 0, stream>>>(pooled, cnt, Wfc, bfc, (float*)d_out);
}